// GAT_14714557956357
// MI455X (gfx1250) — compile-verified
//
#include <hip/hip_runtime.h>

// ---------------------------------------------------------------------------
// Types for V_WMMA_F32_16X16X4_F32 (wave32): A 16x4 f32 = 2 VGPRs/lane,
// B 4x16 f32 = 2 VGPRs/lane, C/D 16x16 f32 = 8 VGPRs/lane.
// ---------------------------------------------------------------------------
typedef __attribute__((ext_vector_type(2))) float v2f;
typedef __attribute__((ext_vector_type(8))) float v8f;

static inline int cdiv_ll(long long a, int b) { return (int)((a + b - 1) / b); }

__device__ __forceinline__ float eluf(float x) {
  return x > 0.f ? x : (__expf(x) - 1.f);
}

// float atomic-max via sign-split monotonic int/uint trick (race-safe).
__device__ __forceinline__ void atomicMaxFloat(float* addr, float value) {
  if (value >= 0.f) {
    atomicMax((int*)addr, __float_as_int(value));
  } else {
    atomicMin((unsigned int*)addr, __float_as_uint(value));
  }
}

// ---------------------------------------------------------------------------
// WMMA f32 GEMM: Out[N,Fout] (+)= A[N,Fin] * W[Fin,Fout]
// One wave computes a 16-row x Fout tile; 8 waves (256 thr) per block = 128
// rows per block. W is staged in LDS (<= 64 KB).
// ---------------------------------------------------------------------------
template <int Fin, int Fout, bool ACC>
__launch_bounds__(256)
__global__ void wmma_gemm_kernel(const float* __restrict__ A,
                                 const float* __restrict__ W,
                                 float* __restrict__ Out, int N) {
  __shared__ float sW[Fin * Fout];
  const int tid = threadIdx.x;
  for (int i = tid; i < Fin * Fout; i += 256) sW[i] = W[i];
  __syncthreads();

  const int wave = tid >> 5;
  const int lane = tid & 31;
  const int rowBase = (blockIdx.x * 8 + wave) * 16;
  if (rowBase >= N) return;  // N is a multiple of 16 (100000 = 6250*16)

  const int half = lane >> 4;   // 0: K{0,1} / M 0..7 side, 1: K{2,3} / M 8..15
  const int l16  = lane & 15;
  const int koff = half * 2;

  constexpr int CT = Fout / 16;  // column tiles
  v8f acc[CT];
#pragma unroll
  for (int t = 0; t < CT; ++t) {
    if (ACC) {
#pragma unroll
      for (int r = 0; r < 8; ++r)
        acc[t][r] = Out[(size_t)(rowBase + r + 8 * half) * Fout + t * 16 + l16];
    } else {
#pragma unroll
      for (int r = 0; r < 8; ++r) acc[t][r] = 0.f;
    }
  }

  const float* Arow = A + (size_t)(rowBase + l16) * Fin;
#pragma unroll 1
  for (int k = 0; k < Fin; k += 4) {
    v2f a;
    a.x = Arow[k + koff];
    a.y = Arow[k + koff + 1];
#pragma unroll
    for (int t = 0; t < CT; ++t) {
      v2f b;
      b.x = sW[(k + koff) * Fout + t * 16 + l16];
      b.y = sW[(k + koff + 1) * Fout + t * 16 + l16];
      acc[t] = __builtin_amdgcn_wmma_f32_16x16x4_f32(
          false, a, false, b, (short)0, acc[t], false, false);
    }
  }

#pragma unroll
  for (int t = 0; t < CT; ++t)
#pragma unroll
    for (int r = 0; r < 8; ++r)
      Out[(size_t)(rowBase + r + 8 * half) * Fout + t * 16 + l16] = acc[t][r];
}

// ---------------------------------------------------------------------------
// el[n,h] = sum_d feat[n,h,d]*al[h,d]; er likewise. feat flat: (n*H+h)*32.
// ---------------------------------------------------------------------------
__global__ void attn_coeff_kernel(const float* __restrict__ feat,
                                  const float* __restrict__ al,
                                  const float* __restrict__ ar,
                                  float* __restrict__ el, float* __restrict__ er,
                                  int N, int H) {
  long long idx = (long long)blockIdx.x * blockDim.x + threadIdx.x;
  if (idx >= (long long)N * H) return;
  int h = (int)(idx % H);
  const float* f = feat + (size_t)idx * 32;
  float sl = 0.f, sr = 0.f;
#pragma unroll
  for (int d = 0; d < 32; ++d) {
    float v = f[d];
    sl += v * al[h * 32 + d];
    sr += v * ar[h * 32 + d];
  }
  el[idx] = sl;
  er[idx] = sr;
}

__global__ void fill_kernel(float* __restrict__ p, float v, size_t n) {
  size_t i = (size_t)blockIdx.x * blockDim.x + threadIdx.x;
  if (i < n) p[i] = v;
}

// e = leaky_relu(el[src]+er[dst]); segment max into m[dst].
__global__ void edge_logits_kernel(const int* __restrict__ src,
                                   const int* __restrict__ dst,
                                   const float* __restrict__ el,
                                   const float* __restrict__ er,
                                   float* __restrict__ ebuf,
                                   float* __restrict__ m, int E, int H) {
  long long idx = (long long)blockIdx.x * blockDim.x + threadIdx.x;
  if (idx >= (long long)E * H) return;
  int e = (int)(idx / H);
  int h = (int)(idx - (long long)e * H);
  float v = el[(size_t)src[e] * H + h] + er[(size_t)dst[e] * H + h];
  v = v > 0.f ? v : 0.2f * v;
  ebuf[idx] = v;
  atomicMaxFloat(&m[(size_t)dst[e] * H + h], v);
}

__global__ void edge_denom_kernel(const int* __restrict__ dst,
                                  const float* __restrict__ ebuf,
                                  const float* __restrict__ m,
                                  float* __restrict__ den, int E, int H) {
  long long idx = (long long)blockIdx.x * blockDim.x + threadIdx.x;
  if (idx >= (long long)E * H) return;
  int e = (int)(idx / H);
  int h = (int)(idx - (long long)e * H);
  size_t d = (size_t)dst[e] * H + h;
  atomicAdd(&den[d], __expf(ebuf[idx] - m[d]));
}

__global__ void edge_alpha_kernel(const int* __restrict__ dst,
                                  float* __restrict__ ebuf,
                                  const float* __restrict__ m,
                                  const float* __restrict__ den, int E, int H) {
  long long idx = (long long)blockIdx.x * blockDim.x + threadIdx.x;
  if (idx >= (long long)E * H) return;
  int e = (int)(idx / H);
  int h = (int)(idx - (long long)e * H);
  size_t d = (size_t)dst[e] * H + h;
  ebuf[idx] = __expf(ebuf[idx] - m[d]) / den[d];
}

// rst[dst,c] += alpha[e,h]*feat[src,c]; one thread per (edge, 4 channels).
template <int F>
__global__ void edge_aggregate_kernel(const int* __restrict__ src,
                                      const int* __restrict__ dst,
                                      const float* __restrict__ alpha,
                                      const float* __restrict__ feat,
                                      float* __restrict__ rst, int E, int H) {
  long long idx = (long long)blockIdx.x * blockDim.x + threadIdx.x;
  constexpr int C4 = F / 4;
  if (idx >= (long long)E * C4) return;
  int e = (int)(idx / C4);
  int c = (int)(idx - (long long)e * C4) * 4;
  int h = c >> 5;
  float a = alpha[(size_t)e * H + h];
  const float4 f = *(const float4*)(feat + (size_t)src[e] * F + c);
  float* out = rst + (size_t)dst[e] * F + c;
  atomicAdd(out + 0, a * f.x);
  atomicAdd(out + 1, a * f.y);
  atomicAdd(out + 2, a * f.z);
  atomicAdd(out + 3, a * f.w);
}

// ELU chain + per-row sumsq + per-column sums (LDS-reduced).
template <int F>
__global__ void fin1_kernel(const float* __restrict__ rst, float* __restrict__ y,
                            float* __restrict__ colsum,
                            float* __restrict__ invnorm, int N, int eluCount) {
  __shared__ float ls[F];
  for (int i = threadIdx.x; i < F; i += blockDim.x) ls[i] = 0.f;
  __syncthreads();
  int n = blockIdx.x * blockDim.x + threadIdx.x;
  if (n < N) {
    const float* r = rst + (size_t)n * F;
    float* yy = y + (size_t)n * F;
    float ss = 0.f;
    for (int c = 0; c < F; ++c) {
      float v = eluf(r[c]);
      if (eluCount == 2) v = eluf(v);
      ss += v * v;
      yy[c] = v;
      atomicAdd(&ls[c], v);
    }
    invnorm[n] = 1.f / sqrtf(1e-6f + ss);
  }
  __syncthreads();
  for (int i = threadIdx.x; i < F; i += blockDim.x)
    atomicAdd(&colsum[i], ls[i]);
}

template <int F>
__global__ void fin2_kernel(const float* __restrict__ y,
                            const float* __restrict__ colsum,
                            const float* __restrict__ invnorm,
                            float* __restrict__ out, int N) {
  long long idx = (long long)blockIdx.x * blockDim.x + threadIdx.x;
  if (idx >= (long long)N * F) return;
  int n = (int)(idx / F);
  int c = (int)(idx - (long long)n * F);
  out[idx] = y[idx] * invnorm[n] - colsum[c] * (1.0f / N);
}

// Final readout: out[c] = mean_n rst[n,c]  (32 columns).
__global__ void mean_reduce_kernel(const float* __restrict__ rst,
                                   float* __restrict__ out, int N) {
  int tid = blockIdx.x * blockDim.x + threadIdx.x;
  int stride = gridDim.x * blockDim.x;
  float acc[32];
#pragma unroll
  for (int c = 0; c < 32; ++c) acc[c] = 0.f;
  for (int n = tid; n < N; n += stride) {
    const float* r = rst + (size_t)n * 32;
#pragma unroll
    for (int c = 0; c < 32; ++c) acc[c] += r[c];
  }
  float invN = 1.f / (float)N;
#pragma unroll
  for (int c = 0; c < 32; ++c) atomicAdd(&out[c], acc[c] * invN);
}

// ---------------------------------------------------------------------------
// One GAT layer (up to the aggregated+residual rst). PairNorm done by caller.
// ---------------------------------------------------------------------------
template <int Fin, int Fout, int H>
static void run_gat_layer(const float* hin, const float* W, const float* al,
                          const float* ar, const float* resW, const int* src,
                          const int* dst, int N, int E, float* feat, float* rst,
                          float* el, float* er, float* mmax, float* den,
                          float* ebuf, hipStream_t s) {
  wmma_gemm_kernel<Fin, Fout, false><<<cdiv_ll(N, 128), 256, 0, s>>>(hin, W, feat, N);
  attn_coeff_kernel<<<cdiv_ll((long long)N * H, 256), 256, 0, s>>>(feat, al, ar, el, er, N, H);
  fill_kernel<<<cdiv_ll((long long)N * H, 256), 256, 0, s>>>(mmax, -3.0e38f, (size_t)N * H);
  fill_kernel<<<cdiv_ll((long long)N * H, 256), 256, 0, s>>>(den, 0.f, (size_t)N * H);
  fill_kernel<<<cdiv_ll((long long)N * Fout, 256), 256, 0, s>>>(rst, 0.f, (size_t)N * Fout);
  long long EH = (long long)E * H;
  edge_logits_kernel<<<cdiv_ll(EH, 256), 256, 0, s>>>(src, dst, el, er, ebuf, mmax, E, H);
  edge_denom_kernel<<<cdiv_ll(EH, 256), 256, 0, s>>>(dst, ebuf, mmax, den, E, H);
  edge_alpha_kernel<<<cdiv_ll(EH, 256), 256, 0, s>>>(dst, ebuf, mmax, den, E, H);
  edge_aggregate_kernel<Fout>
      <<<cdiv_ll((long long)E * (Fout / 4), 256), 256, 0, s>>>(src, dst, ebuf, feat, rst, E, H);
  if (resW)
    wmma_gemm_kernel<Fin, Fout, true><<<cdiv_ll(N, 128), 256, 0, s>>>(hin, resW, rst, N);
}

extern "C" void kernel_launch(void* const* d_in, const int* in_sizes, int n_in,
                              void* d_out, int out_size, void* d_ws,
                              size_t ws_size, hipStream_t stream) {
  const float* x     = (const float*)d_in[0];
  const int*   src   = (const int*)d_in[1];
  const int*   dst   = (const int*)d_in[2];
  const float* W0    = (const float*)d_in[3];
  const float* al0   = (const float*)d_in[4];
  const float* ar0   = (const float*)d_in[5];
  const float* W1    = (const float*)d_in[6];
  const float* al1   = (const float*)d_in[7];
  const float* ar1   = (const float*)d_in[8];
  const float* resW1 = (const float*)d_in[9];
  const float* W2    = (const float*)d_in[10];
  const float* al2   = (const float*)d_in[11];
  const float* ar2   = (const float*)d_in[12];
  const float* resW2 = (const float*)d_in[13];

  const int N = in_sizes[0] / 64;
  const int E = in_sizes[1];

  float* p = (float*)d_ws;
  float* hbuf = p;  p += (size_t)N * 128;   // pairnorm output / layer input
  float* feat = p;  p += (size_t)N * 128;   // h@W (also reused as y in fin1/2)
  float* rst  = p;  p += (size_t)N * 128;   // aggregation accumulator
  float* el   = p;  p += (size_t)N * 4;
  float* er   = p;  p += (size_t)N * 4;
  float* mmax = p;  p += (size_t)N * 4;
  float* den  = p;  p += (size_t)N * 4;
  float* inv  = p;  p += (size_t)N;
  float* ebuf = p;  p += (size_t)E * 4;
  float* csum = p;  p += 128;

  // ---- Layer 0: x[N,64] -> rst[N,128], ELU x1, PairNorm -> hbuf ----
  run_gat_layer<64, 128, 4>(x, W0, al0, ar0, nullptr, src, dst, N, E,
                            feat, rst, el, er, mmax, den, ebuf, stream);
  fill_kernel<<<1, 256, 0, stream>>>(csum, 0.f, 128);
  fin1_kernel<128><<<cdiv_ll(N, 256), 256, 0, stream>>>(rst, feat, csum, inv, N, 1);
  fin2_kernel<128><<<cdiv_ll((long long)N * 128, 256), 256, 0, stream>>>(feat, csum, inv, hbuf, N);

  // ---- Layer 1: hbuf -> rst (+ hbuf@resW1), ELU x2, PairNorm -> hbuf ----
  run_gat_layer<128, 128, 4>(hbuf, W1, al1, ar1, resW1, src, dst, N, E,
                             feat, rst, el, er, mmax, den, ebuf, stream);
  fill_kernel<<<1, 256, 0, stream>>>(csum, 0.f, 128);
  fin1_kernel<128><<<cdiv_ll(N, 256), 256, 0, stream>>>(rst, feat, csum, inv, N, 2);
  fin2_kernel<128><<<cdiv_ll((long long)N * 128, 256), 256, 0, stream>>>(feat, csum, inv, hbuf, N);

  // ---- Layer 2: hbuf -> rst[N,32] (+ hbuf@resW2); mean over nodes ----
  run_gat_layer<128, 32, 1>(hbuf, W2, al2, ar2, resW2, src, dst, N, E,
                            feat, rst, el, er, mmax, den, ebuf, stream);
  fill_kernel<<<1, 256, 0, stream>>>((float*)d_out, 0.f, 32);
  mean_reduce_kernel<<<256, 256, 0, stream>>>(rst, (float*)d_out, N);
}